// SMoE_49941879718237
// MI455X (gfx1250) — compile-verified
//
#include <hip/hip_runtime.h>
#include <math.h>

typedef __attribute__((ext_vector_type(2))) float v2f;
typedef __attribute__((ext_vector_type(8))) float v8f;

#define T_TOK 8192
#define Dd 1024
#define Ff 4096
#define Ee 8
#define MT 32            // tokens per tile in the FFN kernel
#define XP 1028          // LDS pitch for X tile (bank-conflict-free: stride 4 banks/row)
#define HP 68            // LDS pitch for h tile
#define FCHUNK 64        // F columns produced per chunk (8 waves -> gemm1: 2M x 4N of 16)

// ---------------- Kernel 1: RMSNorm + gating + top-2 softmax ----------------
__global__ __launch_bounds__(256) void k_rms_gate(
    const float* __restrict__ x, const float* __restrict__ ln_w,
    const float* __restrict__ Wg, const float* __restrict__ bg,
    float* __restrict__ xn, int* __restrict__ gate_idx, float* __restrict__ gate_w)
{
    const int t   = blockIdx.x;
    const int tid = threadIdx.x;
    const int lane = tid & 31, wid = tid >> 5;
    __shared__ float sRed[8];
    __shared__ float sPart[8 * 8];
    __shared__ float sLog[8];
    __shared__ float sRms;

    const float4 xv = *(const float4*)(x + (size_t)t * Dd + tid * 4);
    float ss = xv.x * xv.x + xv.y * xv.y + xv.z * xv.z + xv.w * xv.w;
    #pragma unroll
    for (int off = 16; off > 0; off >>= 1) ss += __shfl_down(ss, off, 32);
    if (lane == 0) sRed[wid] = ss;
    __syncthreads();
    if (tid == 0) {
        float s = 0.0f;
        #pragma unroll
        for (int i = 0; i < 8; i++) s += sRed[i];
        // eps = finfo(float32).eps = 2^-23
        sRms = 1.0f / sqrtf(s * (1.0f / (float)Dd) + 1.1920928955078125e-07f);
    }
    __syncthreads();
    const float r = sRms;
    const float4 lw = *(const float4*)(ln_w + tid * 4);
    float4 xnv;
    xnv.x = xv.x * r * lw.x;
    xnv.y = xv.y * r * lw.y;
    xnv.z = xv.z * r * lw.z;
    xnv.w = xv.w * r * lw.w;
    *(float4*)(xn + (size_t)t * Dd + tid * 4) = xnv;

    // gating partials: logits = xn @ Wg + bg, Wg is [D,E] row-major
    float part[8];
    #pragma unroll
    for (int e2 = 0; e2 < 8; e2++) part[e2] = 0.0f;
    const float xs[4] = { xnv.x, xnv.y, xnv.z, xnv.w };
    #pragma unroll
    for (int j = 0; j < 4; j++) {
        const float* wr = Wg + (size_t)(tid * 4 + j) * Ee;
        #pragma unroll
        for (int e2 = 0; e2 < 8; e2++) part[e2] += xs[j] * wr[e2];
    }
    #pragma unroll
    for (int e2 = 0; e2 < 8; e2++) {
        float v = part[e2];
        #pragma unroll
        for (int off = 16; off > 0; off >>= 1) v += __shfl_down(v, off, 32);
        if (lane == 0) sPart[wid * 8 + e2] = v;
    }
    __syncthreads();
    if (tid < 8) {
        float s = bg[tid];
        #pragma unroll
        for (int w = 0; w < 8; w++) s += sPart[w * 8 + tid];
        sLog[tid] = s;
    }
    __syncthreads();
    if (tid == 0) {
        int i0 = 0; float v0 = sLog[0];
        for (int e2 = 1; e2 < 8; e2++) { if (sLog[e2] > v0) { v0 = sLog[e2]; i0 = e2; } }
        int i1 = (i0 == 0) ? 1 : 0; float v1 = sLog[i1];
        for (int e2 = 0; e2 < 8; e2++) {
            if (e2 != i0 && sLog[e2] > v1) { v1 = sLog[e2]; i1 = e2; }
        }
        const float d  = __expf(v1 - v0);          // v1 <= v0, stable
        const float w0 = 1.0f / (1.0f + d);
        gate_idx[2 * t] = i0; gate_idx[2 * t + 1] = i1;
        gate_w[2 * t] = w0;   gate_w[2 * t + 1] = 1.0f - w0;
    }
}

// ---------------- Kernel 2: deterministic per-expert compaction ----------------
__global__ __launch_bounds__(256) void k_compact(
    const int* __restrict__ gate_idx, const float* __restrict__ gate_w,
    int* __restrict__ exp_tok, float* __restrict__ exp_wt, int* __restrict__ exp_cnt)
{
    const int e   = blockIdx.x;
    const int tid = threadIdx.x;
    __shared__ int sscan[256];
    __shared__ int sBase;
    if (tid == 0) sBase = 0;
    __syncthreads();
    for (int t0 = 0; t0 < T_TOK; t0 += 256) {
        const int t  = t0 + tid;
        const int i0 = gate_idx[2 * t];
        const int i1 = gate_idx[2 * t + 1];
        const int pred = (i0 == e) || (i1 == e);
        const float w  = (i0 == e) ? gate_w[2 * t] : gate_w[2 * t + 1];
        sscan[tid] = pred;
        __syncthreads();
        for (int off = 1; off < 256; off <<= 1) {
            int v = (tid >= off) ? sscan[tid - off] : 0;
            __syncthreads();
            sscan[tid] += v;
            __syncthreads();
        }
        if (pred) {
            const int pos = sBase + sscan[tid] - 1;
            exp_tok[e * T_TOK + pos] = t;
            exp_wt[e * T_TOK + pos]  = w;
        }
        __syncthreads();
        if (tid == 0) sBase += sscan[255];
        __syncthreads();
    }
    if (tid == 0) exp_cnt[e] = sBase;
}

// ---------------- Kernel 3: fused expert FFN (fp32 WMMA) ----------------
// grid = Ee * (T_TOK/MT) blocks, 256 threads (8 waves).
__global__ __launch_bounds__(256) void k_moe_ffn(
    const float* __restrict__ xn,
    const float* __restrict__ W1a, const float* __restrict__ b1a,
    const float* __restrict__ W1b, const float* __restrict__ b1b,
    const float* __restrict__ W2,  const float* __restrict__ b2,
    const int* __restrict__ exp_tok, const float* __restrict__ exp_wt,
    const int* __restrict__ exp_cnt, float* __restrict__ out)
{
    const int TILES = T_TOK / MT;           // 256
    const int e    = blockIdx.x / TILES;
    const int tile = blockIdx.x % TILES;
    const int cnt  = exp_cnt[e];
    const int m0   = tile * MT;
    if (m0 >= cnt) return;                  // uniform exit: EXEC stays all-ones for WMMA

    __shared__ alignas(16) float sX[MT * XP];   // gathered xn tile, 32 x 1024 (padded)
    __shared__ alignas(16) float sH[MT * HP];   // h chunk, 32 x 64 (padded)
    __shared__ int   sTok[MT];
    __shared__ float sWt[MT];

    const int tid  = threadIdx.x;
    const int lane = tid & 31, wave = tid >> 5;
    const int half = lane >> 4, row = lane & 15;
    const int mi   = wave >> 2;             // 0..1 : which 16-row half of the tile
    const int qi   = wave & 3;              // 0..3 : N-group (gemm1) / 256-col group (gemm2)

    if (tid < MT) {
        const int g  = m0 + tid;
        const int gc = (g < cnt) ? g : (cnt - 1);
        sTok[tid] = exp_tok[e * T_TOK + gc];
        sWt[tid]  = (g < cnt) ? exp_wt[e * T_TOK + gc] : 0.0f;   // padding rows contribute 0
    }
    __syncthreads();

    // Stage X tile: 32 rows x 1024 cols, float4 per thread, coalesced per row.
    {
        const int r  = tid >> 3;
        const int c0 = (tid & 7) * 4;
        const float* src = xn + (size_t)sTok[r] * Dd;
        float* dst = sX + r * XP;
        #pragma unroll
        for (int it = 0; it < Dd / 32; it++) {
            const int c = c0 + it * 32;
            *(float4*)(dst + c) = *(const float4*)(src + c);
        }
    }
    __syncthreads();

    v8f y[16] = {};                          // 16 accum tiles: 16 rows x 256 cols per wave

    const size_t w1base = (size_t)e * Dd * Ff;
    const float* paBase = W1a + w1base + (size_t)(2 * half) * Ff;
    const float* pbBase = W1b + w1base + (size_t)(2 * half) * Ff;

    for (int f0 = 0; f0 < Ff; f0 += FCHUNK) {
        // ---- GEMM1: h[32][64] = silu(X@W1a + b1a) * (X@W1b + b1b) for this chunk ----
        const int ncol = f0 + qi * 16 + row;            // global F column for this lane
        v8f acc_a = {}; v8f acc_b = {};
        const float* pa = paBase + ncol;
        const float* pb = pbBase + ncol;
        const float* px = sX + (mi * 16 + row) * XP + 2 * half;
        #pragma unroll 4
        for (int k0 = 0; k0 < Dd; k0 += 4) {
            v2f a = *(const v2f*)(px + k0);             // A: lanes 0-15 K=k0,k0+1; 16-31 K=k0+2,k0+3
            v2f ba; ba.x = pa[0]; ba.y = pa[Ff];        // B: rows k, k+1 (per half), same N col
            v2f bb; bb.x = pb[0]; bb.y = pb[Ff];
            pa += 4 * Ff; pb += 4 * Ff;
            acc_a = __builtin_amdgcn_wmma_f32_16x16x4_f32(false, a, false, ba, (short)0, acc_a, false, false);
            acc_b = __builtin_amdgcn_wmma_f32_16x16x4_f32(false, a, false, bb, (short)0, acc_b, false, false);
        }
        const float b1av = b1a[(size_t)e * Ff + ncol];
        const float b1bv = b1b[(size_t)e * Ff + ncol];
        float* hcol = sH + (qi * 16 + row);
        #pragma unroll
        for (int v = 0; v < 8; v++) {
            const float av = acc_a[v] + b1av;
            const float bv = acc_b[v] + b1bv;
            // silu via fast hardware reciprocal (v_rcp_f32, ~1 ulp) instead of IEEE divide
            const float s  = av * __builtin_amdgcn_rcpf(1.0f + __expf(-av));
            hcol[(mi * 16 + v + 8 * half) * HP] = s * bv;
        }
        __syncthreads();

        // ---- GEMM2: y += h_chunk @ W2[f0:f0+64, qi*256 : qi*256+256] ----
        const float* pw2 = W2 + ((size_t)e * Ff + f0 + 2 * half) * Dd + qi * 256 + row;
        const float* ph  = sH + (mi * 16 + row) * HP + 2 * half;
        for (int kk = 0; kk < FCHUNK; kk += 4) {
            v2f a = *(const v2f*)(ph + kk);
            #pragma unroll
            for (int j = 0; j < 16; j++) {
                v2f b; b.x = pw2[j * 16]; b.y = pw2[j * 16 + Dd];
                y[j] = __builtin_amdgcn_wmma_f32_16x16x4_f32(false, a, false, b, (short)0, y[j], false, false);
            }
            pw2 += 4 * Dd;
        }
        __syncthreads();                                 // protect sH before next chunk's writes
    }

    // ---- Epilogue: add b2, scale by gate weight, scatter-accumulate ----
    int tokv[8]; float wv[8];
    #pragma unroll
    for (int v = 0; v < 8; v++) {
        const int m = mi * 16 + v + 8 * half;
        tokv[v] = sTok[m];
        wv[v]   = sWt[m];
    }
    #pragma unroll
    for (int j = 0; j < 16; j++) {
        const int col  = qi * 256 + j * 16 + row;
        const float b2v = b2[(size_t)e * Dd + col];
        #pragma unroll
        for (int v = 0; v < 8; v++) {
            const float val = (y[j][v] + b2v) * wv[v];
            atomicAdd(out + (size_t)tokv[v] * Dd + col, val);
        }
    }
}

extern "C" void kernel_launch(void* const* d_in, const int* in_sizes, int n_in,
                              void* d_out, int out_size, void* d_ws, size_t ws_size,
                              hipStream_t stream) {
    (void)in_sizes; (void)n_in; (void)ws_size;
    const float* x   = (const float*)d_in[0];
    const float* lnw = (const float*)d_in[1];
    const float* Wg  = (const float*)d_in[2];
    const float* bg  = (const float*)d_in[3];
    const float* W1a = (const float*)d_in[4];
    const float* b1a = (const float*)d_in[5];
    const float* W1b = (const float*)d_in[6];
    const float* b1b = (const float*)d_in[7];
    const float* W2  = (const float*)d_in[8];
    const float* b2  = (const float*)d_in[9];
    float* out = (float*)d_out;

    char* ws = (char*)d_ws;
    float* xn       = (float*)(ws);                         // 32 MB
    int*   gate_idx = (int*)  (ws + 33554432);              // 64 KB
    float* gate_w   = (float*)(ws + 33619968);              // 64 KB
    int*   exp_tok  = (int*)  (ws + 33685504);              // 256 KB
    float* exp_wt   = (float*)(ws + 33947648);              // 256 KB
    int*   exp_cnt  = (int*)  (ws + 34209792);              // 32 B

    hipMemsetAsync(d_out, 0, (size_t)out_size * sizeof(float), stream);
    k_rms_gate<<<T_TOK, 256, 0, stream>>>(x, lnw, Wg, bg, xn, gate_idx, gate_w);
    k_compact<<<Ee, 256, 0, stream>>>(gate_idx, gate_w, exp_tok, exp_wt, exp_cnt);
    k_moe_ffn<<<Ee * (T_TOK / MT), 256, 0, stream>>>(xn, W1a, b1a, W1b, b1b, W2, b2,
                                                     exp_tok, exp_wt, exp_cnt, out);
}